// AFDLoss_31920196944078
// MI455X (gfx1250) — compile-verified
//
#include <hip/hip_runtime.h>
#include <hip/hip_bf16.h>
#include <math.h>

#define NUM_CLASSES 1000
#define FEATURE_DIM 1024
#define BATCH 8192
#define MOMENTUM 0.9f
#define LAMBDA_INTER 0.5f

#define TILES 63            // 1008 / 16
#define PAD_C 1008
#define NPAIRS_F 499500.0f  // 1000*999/2

typedef __attribute__((ext_vector_type(2))) float v2f;
typedef __attribute__((ext_vector_type(8))) float v8f;

// ---- workspace layout (in floats) ----
#define OFF_SUMS   0u
#define OFF_COUNTS (1024u * 1000u)               // 1,024,000
#define OFF_CN     (OFF_COUNTS + 1024u)          // 1,025,024 (16B aligned *4)
#define OFF_SQ     (OFF_CN + 1024u * 1008u)      // 2,057,216
#define OFF_ACC    (OFF_SQ + 1024u)              // 2,058,240 : [0]=intra_c [1]=intra_a [2]=inter
#define ZERO_N     (OFF_COUNTS + 1024u)          // zero sums + counts

__device__ __forceinline__ float block_reduce_sum(float v, float* s) {
    int t = threadIdx.x;
    s[t] = v; __syncthreads();
    for (int off = 128; off > 0; off >>= 1) {
        if (t < off) s[t] += s[t + off];
        __syncthreads();
    }
    float r = s[0];
    __syncthreads();
    return r;
}

__global__ void k_zero(float* ws) {
    unsigned i = blockIdx.x * 256u + threadIdx.x;
    if (i < ZERO_N) ws[OFF_SUMS + i] = 0.0f;
    if (blockIdx.x == 0 && threadIdx.x < 16) ws[OFF_ACC + threadIdx.x] = 0.0f;
}

// Fused: L2-normalize row of features + scatter into per-class sums (segment sum).
__global__ void k_segsum(const float* __restrict__ f, const int* __restrict__ labels,
                         float* __restrict__ ws) {
    __shared__ float s[256];
    const int b = blockIdx.x, t = threadIdx.x;
    const float4 v = ((const float4*)(f + (size_t)b * FEATURE_DIM))[t];
    const float ssq = v.x * v.x + v.y * v.y + v.z * v.z + v.w * v.w;
    const float tot = block_reduce_sum(ssq, s);
    const float sc = 1.0f / fmaxf(sqrtf(tot), 1e-12f);
    const int lab = labels[b];
    float* sums = ws + OFF_SUMS + (size_t)lab * FEATURE_DIM + 4u * t;
    atomicAdd(sums + 0, v.x * sc);
    atomicAdd(sums + 1, v.y * sc);
    atomicAdd(sums + 2, v.z * sc);
    atomicAdd(sums + 3, v.w * sc);
    if (t == 0) atomicAdd(&ws[OFF_COUNTS + lab], 1.0f);
}

// Momentum center update -> padded centers_new [1008 x 1024], plus per-row ||c||^2.
__global__ void k_centers(const float* __restrict__ centers, float* __restrict__ ws) {
    __shared__ float s[256];
    const int c = blockIdx.x, t = threadIdx.x;
    float* cn = ws + OFF_CN + (size_t)c * FEATURE_DIM;
    if (c >= NUM_CLASSES) {               // zero padding rows (uniform per block)
        ((float4*)cn)[t] = make_float4(0.f, 0.f, 0.f, 0.f);
        if (t == 0) ws[OFF_SQ + c] = 0.0f;
        return;
    }
    const float4 cv = ((const float4*)(centers + (size_t)c * FEATURE_DIM))[t];
    // center_is_zero = all(center == 0): max-abs reduction
    float ma = fmaxf(fmaxf(fabsf(cv.x), fabsf(cv.y)), fmaxf(fabsf(cv.z), fabsf(cv.w)));
    s[t] = ma; __syncthreads();
    for (int off = 128; off > 0; off >>= 1) {
        if (t < off) s[t] = fmaxf(s[t], s[t + off]);
        __syncthreads();
    }
    const bool center_zero = (s[0] == 0.0f);
    __syncthreads();

    const float cnt = ws[OFF_COUNTS + c];
    const bool has = (cnt > 0.5f);
    const float inv = 1.0f / fmaxf(cnt, 1.0f);
    const float4 sm = ((const float4*)(ws + OFF_SUMS + (size_t)c * FEATURE_DIM))[t];
    float4 mean = make_float4(sm.x * inv, sm.y * inv, sm.z * inv, sm.w * inv);
    float4 out;
    out.x = has ? (center_zero ? mean.x : MOMENTUM * cv.x + (1.0f - MOMENTUM) * mean.x) : cv.x;
    out.y = has ? (center_zero ? mean.y : MOMENTUM * cv.y + (1.0f - MOMENTUM) * mean.y) : cv.y;
    out.z = has ? (center_zero ? mean.z : MOMENTUM * cv.z + (1.0f - MOMENTUM) * mean.z) : cv.z;
    out.w = has ? (center_zero ? mean.w : MOMENTUM * cv.w + (1.0f - MOMENTUM) * mean.w) : cv.w;
    ((float4*)cn)[t] = out;

    const float sq = out.x * out.x + out.y * out.y + out.z * out.z + out.w * out.w;
    const float tsq = block_reduce_sum(sq, s);
    if (t == 0) ws[OFF_SQ + c] = tsq;
}

// Fused intra losses: normalize f and fa rows, gather center, accumulate mean norms.
__global__ void k_intra(const float* __restrict__ f, const float* __restrict__ fa,
                        const int* __restrict__ labels, float* __restrict__ ws) {
    __shared__ float s[256];
    const int b = blockIdx.x, t = threadIdx.x;
    const float4 vf = ((const float4*)(f  + (size_t)b * FEATURE_DIM))[t];
    const float4 va = ((const float4*)(fa + (size_t)b * FEATURE_DIM))[t];
    const int lab = labels[b];
    const float4 vc = ((const float4*)(ws + OFF_CN + (size_t)lab * FEATURE_DIM))[t];

    const float tf = block_reduce_sum(vf.x*vf.x + vf.y*vf.y + vf.z*vf.z + vf.w*vf.w, s);
    const float ta = block_reduce_sum(va.x*va.x + va.y*va.y + va.z*va.z + va.w*va.w, s);
    const float sf = 1.0f / fmaxf(sqrtf(tf), 1e-12f);
    const float sa = 1.0f / fmaxf(sqrtf(ta), 1e-12f);

    float dc, da, tmp;
    tmp = vf.x*sf - vc.x; dc  = tmp*tmp;
    tmp = vf.y*sf - vc.y; dc += tmp*tmp;
    tmp = vf.z*sf - vc.z; dc += tmp*tmp;
    tmp = vf.w*sf - vc.w; dc += tmp*tmp;
    tmp = va.x*sa - vc.x; da  = tmp*tmp;
    tmp = va.y*sa - vc.y; da += tmp*tmp;
    tmp = va.z*sa - vc.z; da += tmp*tmp;
    tmp = va.w*sa - vc.w; da += tmp*tmp;

    const float tdc = block_reduce_sum(dc, s);
    const float tda = block_reduce_sum(da, s);
    if (t == 0) {
        atomicAdd(&ws[OFF_ACC + 0], sqrtf(tdc) * (1.0f / BATCH));
        atomicAdd(&ws[OFF_ACC + 1], sqrtf(tda) * (1.0f / BATCH));
    }
}

// Gram tiles via V_WMMA_F32_16X16X4_F32; one 16x16 tile per wave over the upper triangle.
// Inter-loss applied in-register, never materializing G.
__global__ void k_gram(float* __restrict__ ws) {
    const float* __restrict__ cn = ws + OFF_CN;
    const float* __restrict__ sq = ws + OFF_SQ;
    const int wave = threadIdx.x >> 5;
    const int lane = threadIdx.x & 31;
    const int p = blockIdx.x * 8 + wave;               // 252 blocks * 8 waves = 2016 pairs
    const int NP = TILES * (TILES + 1) / 2;
    if (p >= NP) return;                                // wave-uniform branch: EXEC stays all-1s

    int i = 0, rem = p;                                 // decode upper-triangle pair (i <= j)
    while (rem >= (TILES - i)) { rem -= (TILES - i); ++i; }
    const int j = i + rem;

    const int m    = lane & 15;                         // row within tile (A) / col within tile (B)
    const int koff = (lane >> 4) << 1;                  // lanes 16-31 carry K=2,3 of each step
    const float* rowA = cn + (size_t)(i * 16 + m) * FEATURE_DIM + koff;
    const float* rowB = cn + (size_t)(j * 16 + m) * FEATURE_DIM + koff;

    v8f acc = {0.f, 0.f, 0.f, 0.f, 0.f, 0.f, 0.f, 0.f};
    for (int k = 0; k < FEATURE_DIM; k += 16) {
#pragma unroll
        for (int kk = 0; kk < 16; kk += 4) {
            v2f a = *(const v2f*)(rowA + k + kk);       // global_load_b64, 8B aligned
            v2f b = *(const v2f*)(rowB + k + kk);
            acc = __builtin_amdgcn_wmma_f32_16x16x4_f32(
                false, a, false, b, (short)0, acc, false, false);
        }
    }

    // C/D layout: VGPR v -> M = v + (lane>=16)*8 ; N = lane&15
    const int rbase = i * 16 + ((lane >> 4) << 3);
    const int c = j * 16 + m;
    float contrib = 0.0f;
#pragma unroll
    for (int v = 0; v < 8; ++v) {
        const int r = rbase + v;
        if (r < c && c < NUM_CLASSES) {
            const float d2 = sq[r] + sq[c] - 2.0f * acc[v];
            const float dist = sqrtf(fmaxf(d2, 0.0f));
            contrib += fmaxf(1.0f - dist, 0.0f);
        }
    }
    for (int off = 16; off > 0; off >>= 1)
        contrib += __shfl_xor(contrib, off, 32);
    if (lane == 0) atomicAdd(&ws[OFF_ACC + 2], contrib);
}

__global__ void k_final(const float* __restrict__ ws, float* __restrict__ out) {
    if (threadIdx.x == 0 && blockIdx.x == 0) {
        out[0] = (ws[OFF_ACC + 0] + ws[OFF_ACC + 1])
               - LAMBDA_INTER * ws[OFF_ACC + 2] * (1.0f / NPAIRS_F);
    }
}

extern "C" void kernel_launch(void* const* d_in, const int* in_sizes, int n_in,
                              void* d_out, int out_size, void* d_ws, size_t ws_size,
                              hipStream_t stream) {
    const float* features     = (const float*)d_in[0];  // [8192,1024] f32
    const float* features_adv = (const float*)d_in[1];  // [8192,1024] f32
    const float* centers      = (const float*)d_in[2];  // [1000,1024] f32
    const int*   labels       = (const int*)d_in[3];    // [8192] int32 (JAX default demotes int64)
    float* ws  = (float*)d_ws;
    float* out = (float*)d_out;

    (void)in_sizes; (void)n_in; (void)out_size; (void)ws_size;

    k_zero   <<<ZERO_N / 256, 256, 0, stream>>>(ws);
    k_segsum <<<BATCH, 256, 0, stream>>>(features, labels, ws);
    k_centers<<<PAD_C, 256, 0, stream>>>(centers, ws);
    k_intra  <<<BATCH, 256, 0, stream>>>(features, features_adv, labels, ws);
    k_gram   <<<(TILES * (TILES + 1) / 2) / 8, 256, 0, stream>>>(ws);
    k_final  <<<1, 32, 0, stream>>>(ws, out);
}